// ForwardModel_52123723104730
// MI455X (gfx1250) — compile-verified
//
#include <hip/hip_runtime.h>
#include <math.h>

typedef __attribute__((ext_vector_type(2))) float v2f;
typedef __attribute__((ext_vector_type(8))) float v8f;

#define PI_F 3.14159265358979323846f
#define S68  68          // padded LDS row stride (floats) to dodge bank conflicts
#define PLANE (64 * S68) // 4352 floats per 64x64 plane

// complex tile MAC: (Cr+iCi) += (Ar+iAi)(Br+iBi)
// V_WMMA_F32_16X16X4_F32 : D = A(16x4) * B(4x16) + C(16x16), all f32.
// f32 WMMA has no A/B negate modifier (NEG[1:0]==0 required) -> negate the
// Ai fragment in VALU for the -Ai*Bi term.
__device__ __forceinline__ void cwmma(v8f& cr, v8f& ci, v2f ar, v2f ai, v2f br, v2f bi) {
  v2f nai; nai.x = -ai.x; nai.y = -ai.y;
  cr = __builtin_amdgcn_wmma_f32_16x16x4_f32(false, ar,  false, br, (short)0, cr, false, false);
  cr = __builtin_amdgcn_wmma_f32_16x16x4_f32(false, nai, false, bi, (short)0, cr, false, false);
  ci = __builtin_amdgcn_wmma_f32_16x16x4_f32(false, ar,  false, bi, (short)0, ci, false, false);
  ci = __builtin_amdgcn_wmma_f32_16x16x4_f32(false, ai,  false, br, (short)0, ci, false, false);
}

__global__ void __launch_bounds__(512)
ptycho_fwd_kernel(const float* __restrict__ xr_g, const float* __restrict__ xi_g,
                  const float* __restrict__ pos_g,
                  const float* __restrict__ pbr_g, const float* __restrict__ pbi_g,
                  const float* __restrict__ osf_g, float* __restrict__ out_g) {
  extern __shared__ float smem[];
  float*  XR = smem;               // 4 planes
  float*  XI = XR + 4 * PLANE;     // 4 planes
  float*  WR = XI + 4 * PLANE;
  float*  WI = WR + PLANE;
  float*  TR = WI + PLANE;
  float*  TI = TR + PLANE;
  float*  UR = TI + PLANE;
  float*  UI = UR + PLANE;
  float2* DY = (float2*)(UI + PLANE);   // [16][127]
  float2* DX = DY + 16 * 127;           // [16][127]
  float*  sPos = (float*)(DX + 16 * 127); // 8

  const int b    = blockIdx.x;
  const int tid  = threadIdx.x;
  const int lane = tid & 31;
  const int w    = tid >> 5;       // 16 waves
  const int mt   = w >> 2, nt = w & 3;
  const int half = lane >> 4;      // K-half select for 16x16x4 frags
  const int l16  = lane & 15;
  const int rowB = mt * 16, colB = nt * 16;

  if (tid < 8) sPos[tid] = pos_g[b * 8 + tid];   // (dy,dx) per channel

  // stage x[b, 0..3] (real+imag) into LDS planes
  const float* gr = xr_g + (size_t)b * 16384;
  const float* gi = xi_g + (size_t)b * 16384;
  for (int idx = tid; idx < 16384; idx += 512) {
    int cc = idx >> 12, rem = idx & 4095;
    int row = rem >> 6, col = rem & 63;
    int o = cc * PLANE + row * S68 + col;
    XR[o] = gr[idx];
    XI[o] = gi[idx];
  }
  // W64 DFT matrix: W[j][k] = exp(-2*pi*i*j*k/64)
  for (int idx = tid; idx < 4096; idx += 512) {
    int j = idx >> 6, k = idx & 63;
    float ang = -(2.0f * PI_F / 64.0f) * (float)((j * k) & 63);
    float s, c;
    sincosf(ang, &s, &c);
    WR[j * S68 + k] = c;
    WI[j * S68 + k] = s;
  }
  __syncthreads();

  // Dirichlet shift vectors: D68(t) = e^{-i*pi*t/68} * sin(pi t)/(68 sin(pi t/68))
  // DY/DX[p=c*4+c'][tt] covers m-p = tt-63 in [-63,63]
  for (int idx = tid; idx < 2 * 16 * 127; idx += 512) {
    int dim = idx / 2032;
    int r2  = idx - dim * 2032;
    int p   = r2 / 127;
    int tt  = r2 - p * 127;
    int cA = p >> 2, cB = p & 3;
    float delta = sPos[cB * 2 + dim] - sPos[cA * 2 + dim]; // d_{c'} - d_c
    float t  = (float)(tt - 63) - delta;
    float th = PI_F * t * (1.0f / 68.0f);
    float s1 = sinf(PI_F * t);
    float s2 = sinf(th);
    float rr = (fabsf(t) > 1e-6f) ? s1 / (68.0f * s2) : 1.0f;
    float2 v;
    v.x = rr * cosf(th);
    v.y = -s1 * (1.0f / 68.0f);   // == -rr*sin(th)
    if (dim) DX[p * 127 + tt] = v; else DY[p * 127 + tt] = v;
  }
  __syncthreads();

  const float outScale = 1.0f / (64.0f * osf_g[0]);

  for (int c = 0; c < 4; ++c) {
    v8f accR = {0, 0, 0, 0, 0, 0, 0, 0};   // patch tile accumulators (over c')
    v8f accI = {0, 0, 0, 0, 0, 0, 0, 0};

    for (int cp = 0; cp < 4; ++cp) {
      const float2* dY  = DY + (c * 4 + cp) * 127;
      const float2* dX  = DX + (c * 4 + cp) * 127;
      const float*  xrp = XR + cp * PLANE;
      const float*  xip = XI + cp * PLANE;

      // ---- T = My(c,cp) * x[cp]   (A from Dirichlet vector, B strided plane)
      v8f tr = {0, 0, 0, 0, 0, 0, 0, 0};
      v8f ti = {0, 0, 0, 0, 0, 0, 0, 0};
      for (int kk = 0; kk < 16; ++kk) {
        int k0  = kk * 4 + half * 2;
        int row = rowB + l16;
        int t0  = row - k0 + 63;                 // in [1,126]
        float2 d0 = dY[t0], d1 = dY[t0 - 1];
        v2f ar; ar.x = d0.x; ar.y = d1.x;
        v2f ai; ai.x = d0.y; ai.y = d1.y;
        int n = colB + l16;
        v2f br; br.x = xrp[k0 * S68 + n]; br.y = xrp[(k0 + 1) * S68 + n];
        v2f bi; bi.x = xip[k0 * S68 + n]; bi.y = xip[(k0 + 1) * S68 + n];
        cwmma(tr, ti, ar, ai, br, bi);
      }
      __syncthreads();           // prior readers of T are done
#pragma unroll
      for (int r = 0; r < 8; ++r) {
        int M = rowB + half * 8 + r;
        int N = colB + l16;
        TR[M * S68 + N] = tr[r];
        TI[M * S68 + N] = ti[r];
      }
      __syncthreads();

      // ---- acc += T * Mx(c,cp)^T   (A contiguous plane, B from Dirichlet vector)
      for (int kk = 0; kk < 16; ++kk) {
        int k0  = kk * 4 + half * 2;
        int row = rowB + l16;
        v2f ar; ar.x = TR[row * S68 + k0]; ar.y = TR[row * S68 + k0 + 1];
        v2f ai; ai.x = TI[row * S68 + k0]; ai.y = TI[row * S68 + k0 + 1];
        int n  = colB + l16;
        int t0 = n - k0 + 63;
        float2 d0 = dX[t0], d1 = dX[t0 - 1];
        v2f br; br.x = d0.x; br.y = d1.x;
        v2f bi; bi.x = d0.y; bi.y = d1.y;
        cwmma(accR, accI, ar, ai, br, bi);
      }
    }
    __syncthreads();

    // ---- IL = patch * probe  -> reuse T planes
#pragma unroll
    for (int r = 0; r < 8; ++r) {
      int M = rowB + half * 8 + r;
      int N = colB + l16;
      float pr = pbr_g[M * 64 + N], pim = pbi_g[M * 64 + N];
      float vr = accR[r], vi = accI[r];
      TR[M * S68 + N] = vr * pr - vi * pim;
      TI[M * S68 + N] = vr * pim + vi * pr;
    }
    __syncthreads();

    // ---- U = W64 * IL
    v8f ur = {0, 0, 0, 0, 0, 0, 0, 0};
    v8f ui = {0, 0, 0, 0, 0, 0, 0, 0};
    for (int kk = 0; kk < 16; ++kk) {
      int k0  = kk * 4 + half * 2;
      int row = rowB + l16;
      v2f ar; ar.x = WR[row * S68 + k0]; ar.y = WR[row * S68 + k0 + 1];
      v2f ai; ai.x = WI[row * S68 + k0]; ai.y = WI[row * S68 + k0 + 1];
      int n = colB + l16;
      v2f br; br.x = TR[k0 * S68 + n]; br.y = TR[(k0 + 1) * S68 + n];
      v2f bi; bi.x = TI[k0 * S68 + n]; bi.y = TI[(k0 + 1) * S68 + n];
      cwmma(ur, ui, ar, ai, br, bi);
    }
#pragma unroll
    for (int r = 0; r < 8; ++r) {
      int M = rowB + half * 8 + r;
      int N = colB + l16;
      UR[M * S68 + N] = ur[r];
      UI[M * S68 + N] = ui[r];
    }
    __syncthreads();

    // ---- Xf = U * W64^T ; emit amp with fftshift (index XOR 32)
    v8f fr = {0, 0, 0, 0, 0, 0, 0, 0};
    v8f fi = {0, 0, 0, 0, 0, 0, 0, 0};
    for (int kk = 0; kk < 16; ++kk) {
      int k0  = kk * 4 + half * 2;
      int row = rowB + l16;
      v2f ar; ar.x = UR[row * S68 + k0]; ar.y = UR[row * S68 + k0 + 1];
      v2f ai; ai.x = UI[row * S68 + k0]; ai.y = UI[row * S68 + k0 + 1];
      int n = colB + l16;   // B[k][n] = W[n][k]
      v2f br; br.x = WR[n * S68 + k0]; br.y = WR[n * S68 + k0 + 1];
      v2f bi; bi.x = WI[n * S68 + k0]; bi.y = WI[n * S68 + k0 + 1];
      cwmma(fr, fi, ar, ai, br, bi);
    }
    float* go = out_g + ((size_t)b * 4 + c) * 4096;
#pragma unroll
    for (int r = 0; r < 8; ++r) {
      int M = rowB + half * 8 + r;
      int N = colB + l16;
      float a = sqrtf(fr[r] * fr[r] + fi[r] * fi[r]) * outScale;
      go[((M ^ 32) << 6) | (N ^ 32)] = a;
    }
    __syncthreads();
  }
}

extern "C" void kernel_launch(void* const* d_in, const int* in_sizes, int n_in,
                              void* d_out, int out_size, void* d_ws, size_t ws_size,
                              hipStream_t stream) {
  (void)in_sizes; (void)n_in; (void)d_ws; (void)ws_size; (void)out_size;
  const float* x_real     = (const float*)d_in[0];
  const float* x_imag     = (const float*)d_in[1];
  const float* positions  = (const float*)d_in[2];
  const float* probe_real = (const float*)d_in[3];
  const float* probe_imag = (const float*)d_in[4];
  const float* osf        = (const float*)d_in[5];
  float* out = (float*)d_out;

  const size_t shmem = (size_t)(14 * PLANE + 2 * 2 * 16 * 127 + 8) * sizeof(float); // ~270 KB

  static bool attr_done = false;
  if (!attr_done) {  // host-side attribute set (not a stream op; capture-safe, deterministic)
    (void)hipFuncSetAttribute((const void*)ptycho_fwd_kernel,
                              hipFuncAttributeMaxDynamicSharedMemorySize, (int)shmem);
    attr_done = true;
  }

  ptycho_fwd_kernel<<<dim3(2048), dim3(512), shmem, stream>>>(
      x_real, x_imag, positions, probe_real, probe_imag, osf, out);
}